// ScaledDotProductAttention_34196529611205
// MI455X (gfx1250) — compile-verified
//
#include <hip/hip_runtime.h>
#include <math.h>

// ---------------------------------------------------------------------------
// ScaledDotProductAttention with round+dedup normalizer, CDNA5 (gfx1250)
//   B=2 H=8 N=2048 D=64, fp32 WMMA (V_WMMA_F32_16X16X4_F32) for both GEMMs.
//   One workgroup = 4 wave32 = 128 threads, handles one 16-row Q tile of one
//   (b,h) head. Score panel (16 x 2048 f32) lives in the 320KB WGP LDS.
// ---------------------------------------------------------------------------

#define B_ 2
#define H_ 8
#define N_ 2048
#define D_ 64
#define TILE 16
#define NT (N_ / TILE)            // 128 row tiles per head
#define NWAVES 4
#define NTHREADS (NWAVES * 32)
#define COLS_PER_WAVE (N_ / NWAVES)   // 512
#define SS 2052                   // padded LDS row stride in floats (2048+4):
                                  //  - even (b64-aligned A-frag loads)
                                  //  - mult of 4 (b128-aligned float4 IO)
                                  //  - 8*SS % 64 == 32 -> no half-wave bank clash
#define BMW 3128                  // bitmap words >= ceil(100001/32)=3126, padded

typedef __attribute__((ext_vector_type(2))) float v2f;
typedef __attribute__((ext_vector_type(8))) float v8f;

__device__ __forceinline__ v8f wmma_f32_4(v2f a, v2f b, v8f c) {
  // D(16x16,f32) = A(16x4,f32) * B(4x16,f32) + C
  return __builtin_amdgcn_wmma_f32_16x16x4_f32(
      /*neg_a=*/false, a, /*neg_b=*/false, b,
      /*c_mod=*/(short)0, c, /*reuse_a=*/false, /*reuse_b=*/false);
}

__global__ void ScaledDotProductAttention_34196529611205_kernel(
    const float* __restrict__ q, const float* __restrict__ k,
    const float* __restrict__ v, float* __restrict__ out,
    float* __restrict__ attn) {
  extern __shared__ float smem[];
  float*    sc     = smem;                      // [16][SS] scores -> rounded e (*1e5)
  float*    rowmax = sc + TILE * SS;            // [16]
  float*    pmax   = rowmax + TILE;             // [16][8] partial maxima
  float*    dinv   = pmax + TILE * 8;           // [16] 1/denom_int
  int*      dint   = (int*)(dinv + TILE);       // [16] exact distinct-sum (int)
  unsigned* bitmap = (unsigned*)(dint + TILE);  // [BMW] presence bitmap

  const int tid  = threadIdx.x;
  const int wave = tid >> 5;
  const int lane = tid & 31;
  const int lo   = lane & 15;   // WMMA: M (A row) / N (B,C,D col) index
  const int hi   = lane >> 4;   // WMMA: lane-half selector

  const int wg      = blockIdx.x;
  const int tileIdx = wg & (NT - 1);   // NT == 128
  const int bh      = wg >> 7;
  const int row0    = tileIdx * TILE;

  const float* qh = q + (size_t)bh * N_ * D_;
  const float* kh = k + (size_t)bh * N_ * D_;
  const float* vh = v + (size_t)bh * N_ * D_;

  // ---- Phase 1: preload Q A-fragments (scaled by 1/T), whole d=64 ----------
  // A(16x4) layout: vgpr0 <-> K=2*hi, vgpr1 <-> K=2*hi+1, M = lane&15.
  v2f afrag[16];
  {
    const float* qrow = qh + (size_t)(row0 + lo) * D_ + 2 * hi;
    const float qs = 1.0f / 8.0f;  // 1/sqrt(d_k)
#pragma unroll
    for (int c = 0; c < 16; ++c) {
      float2 t2 = *(const float2*)(qrow + 4 * c);
      v2f a; a.x = t2.x * qs; a.y = t2.y * qs;
      afrag[c] = a;
    }
  }

  // ---- Phase 2: S = (Q/T) @ K^T ; each wave owns 512 columns ---------------
  for (int t = 0; t < COLS_PER_WAVE / 16; ++t) {
    const int col0 = wave * COLS_PER_WAVE + t * 16;
    v8f acc = {0.f, 0.f, 0.f, 0.f, 0.f, 0.f, 0.f, 0.f};
    const float* krow = kh + (size_t)(col0 + lo) * D_ + 2 * hi;
#pragma unroll
    for (int c = 0; c < 16; ++c) {
      float2 t2 = *(const float2*)(krow + 4 * c);
      v2f b; b.x = t2.x; b.y = t2.y;
      acc = wmma_f32_4(afrag[c], b, acc);
    }
    // C/D layout: vgpr r8 <-> row M = r8 + 8*hi, col N = lo
#pragma unroll
    for (int r8 = 0; r8 < 8; ++r8)
      sc[(r8 + 8 * hi) * SS + col0 + lo] = acc[r8];
  }
  __syncthreads();

  // ---- Phase 3: per-row max --------------------------------------------------
  const int r    = tid & 15;     // row this thread scans
  const int part = tid >> 4;     // 8 partials per row
  {
    float* srow = sc + r * SS;
    float mx = -INFINITY;
    const int c0 = part * (N_ / 8);
#pragma unroll 8
    for (int i = 0; i < N_ / 8; ++i) mx = fmaxf(mx, srow[c0 + i]);
    pmax[r * 8 + part] = mx;
  }
  __syncthreads();
  if (tid < TILE) {
    float m = pmax[tid * 8];
#pragma unroll
    for (int p = 1; p < 8; ++p) m = fmaxf(m, pmax[tid * 8 + p]);
    rowmax[tid] = m;
    dint[tid] = 0;
  }
  __syncthreads();

  // ---- Phase 4: e = round(exp(s-m)*1e5), stored as integer-valued float ----
  {
    float* srow = sc + r * SS;
    const float rm = rowmax[r];
    const int c0 = part * (N_ / 8);
#pragma unroll 4
    for (int i = 0; i < N_ / 8; ++i) {
      const int c = c0 + i;
      float ev = rintf(__expf(srow[c] - rm) * 100000.0f);  // RTE == jnp.round
      srow[c] = ev;   // integer value in [0, 100000]
    }
  }
  __syncthreads();

  // ---- Phase 5: exact distinct-sum per row via LDS bitmap -------------------
  for (int rr = 0; rr < TILE; ++rr) {
    for (int w = tid; w < BMW; w += NTHREADS) bitmap[w] = 0u;
    __syncthreads();
    const float* er = sc + rr * SS;
    for (int j = tid; j < N_; j += NTHREADS) {
      const int vi = (int)er[j];
      const unsigned bit = 1u << (vi & 31);
      const unsigned old = atomicOr(&bitmap[vi >> 5], bit);  // ds_or_rtn_b32
      if (!(old & bit)) atomicAdd(&dint[rr], vi);            // first occurrence
    }
    __syncthreads();
  }
  if (tid < TILE) dinv[tid] = 1.0f / (float)dint[tid];  // p = ev / denom_int
  __syncthreads();

  // ---- Phase 6: stream attn_p to HBM (dominant traffic, float4 coalesced) --
  for (int rr = 0; rr < TILE; ++rr) {
    const float4* s4 = (const float4*)(sc + rr * SS);
    const float inv = dinv[rr];
    float4* g4 = (float4*)(attn + ((size_t)(bh * N_ + row0 + rr)) * N_);
#pragma unroll
    for (int i = tid; i < N_ / 4; i += NTHREADS) {
      float4 x = s4[i];
      x.x *= inv; x.y *= inv; x.z *= inv; x.w *= inv;
      g4[i] = x;
    }
  }

  // ---- Phase 7: O = (E @ V) * (1/denom); each wave owns one 16-wide d-slice -
  {
    v8f acc = {0.f, 0.f, 0.f, 0.f, 0.f, 0.f, 0.f, 0.f};
    const float* erow  = sc + lo * SS + 2 * hi;       // A: M=lo, K=2*hi(+1)
    const float* vbase = vh + 16 * wave + lo;         // B: N=lo, rows j+2*hi(+1)
#pragma unroll 4
    for (int j = 0; j < N_; j += 4) {
      float2 ea = *(const float2*)(erow + j);         // ds_load_b64
      v2f a; a.x = ea.x; a.y = ea.y;
      v2f b;
      b.x = vbase[(size_t)(j + 2 * hi) * D_];
      b.y = vbase[(size_t)(j + 2 * hi + 1) * D_];
      acc = wmma_f32_4(a, b, acc);
    }
#pragma unroll
    for (int r8 = 0; r8 < 8; ++r8) {
      const int m = r8 + 8 * hi;
      out[((size_t)(bh * N_ + row0 + m)) * D_ + 16 * wave + lo] =
          acc[r8] * dinv[m];
    }
  }
}

extern "C" void kernel_launch(void* const* d_in, const int* in_sizes, int n_in,
                              void* d_out, int out_size, void* d_ws, size_t ws_size,
                              hipStream_t stream) {
  (void)in_sizes; (void)n_in; (void)out_size; (void)d_ws; (void)ws_size;
  const float* q = (const float*)d_in[0];
  const float* k = (const float*)d_in[1];
  const float* v = (const float*)d_in[2];
  float* out  = (float*)d_out;                          // [B,H,N,D]
  float* attn = out + (size_t)B_ * H_ * N_ * D_;        // [B,H,N,N]

  const size_t smem_bytes =
      (size_t)(TILE * SS + TILE + TILE * 8 + TILE + TILE + BMW) * sizeof(float);
  // ~144.5 KB dynamic LDS: fine on 320KB-per-WGP CDNA5, opt in explicitly.
  (void)hipFuncSetAttribute(
      (const void*)ScaledDotProductAttention_34196529611205_kernel,
      hipFuncAttributeMaxDynamicSharedMemorySize, (int)smem_bytes);

  dim3 grid(B_ * H_ * NT);   // 2048 workgroups
  dim3 block(NTHREADS);      // 128 threads = 4 wave32
  ScaledDotProductAttention_34196529611205_kernel<<<grid, block, smem_bytes, stream>>>(
      q, k, v, out, attn);
}